// NNConvPair_10024453669566
// MI455X (gfx1250) — compile-verified
//
#include <hip/hip_runtime.h>
#include <math.h>

// Problem constants (match reference)
#define N_NODES 25000
#define N_EDGES 100000
#define N_BATCH 1000
#define F_IN    32
#define FE_IN   8
#define H_DIM   16
#define YCOLS   288   // 256 (Y[h,o]) + 16 (b2 term) + 16 (root term)

typedef float v2f __attribute__((ext_vector_type(2)));
typedef float v8f __attribute__((ext_vector_type(8)));

// ---------------------------------------------------------------- utilities
__global__ void fill_zero_f32(float* __restrict__ p, int n) {
  int t = blockIdx.x * blockDim.x + threadIdx.x;
  if (t < n) p[t] = 0.0f;
}

// Pack B matrix (cin x 288): cols 0..255 = W2[h, i*16+o] (c = h*16+o),
// cols 256..271 = b2[i*16+o], cols 272..287 = root[i, o].
__global__ void pack_wbig(const float* __restrict__ W2, const float* __restrict__ b2,
                          const float* __restrict__ root, float* __restrict__ Wbig, int cin) {
  int t = blockIdx.x * blockDim.x + threadIdx.x;
  if (t >= cin * YCOLS) return;
  int i = t / YCOLS, c = t - i * YCOLS;
  float v;
  if (c < 256) { int h = c >> 4, o = c & 15; v = W2[h * (cin * 16) + i * 16 + o]; }
  else if (c < 272) v = b2[i * 16 + (c - 256)];
  else              v = root[i * 16 + (c - 272)];
  Wbig[t] = v;
}

// ------------------------------------------------- WMMA GEMM  Y = X @ Wbig
// X: nrows x cin (f32), Wbig: cin x 288, Y: nrows x 288.
// One wave per 16x16 output tile; K-loop of V_WMMA_F32_16X16X4_F32.
// A(16x4) layout: lane m=L&15, VGPR v holds K = (L>=16?2:0)+v.
// B(4x16) layout: lane n=L&15, VGPR v holds K = (L>=16?2:0)+v.
// C/D(16x16): lane col=L&15, VGPR v holds row M = (L>=16?8:0)+v.
__global__ void wmma_gemm_f32(const float* __restrict__ X, const float* __restrict__ Wbig,
                              float* __restrict__ Y, int nrows, int cin) {
  const int nColTiles = YCOLS / 16;  // 18
  int wave = blockIdx.x * (blockDim.x >> 5) + (threadIdx.x >> 5);
  int rowTiles = (nrows + 15) >> 4;
  if (wave >= rowTiles * nColTiles) return;   // wave-uniform: EXEC stays all-1s
  int rowTile = wave / nColTiles;
  int colTile = wave - rowTile * nColTiles;
  int lane = threadIdx.x & 31;
  int mr = lane & 15;
  int kh = (lane >> 4) << 1;                  // 0 or 2
  int row = rowTile * 16 + mr;
  if (row >= nrows) row = nrows - 1;          // clamp reads; stores guarded below
  int col = colTile * 16 + mr;
  const float* xr = X + (size_t)row * cin;
  v8f acc = {};
  for (int k0 = 0; k0 < cin; k0 += 4) {
    v2f a, b;
    a.x = xr[k0 + kh];
    a.y = xr[k0 + kh + 1];
    b.x = Wbig[(k0 + kh) * YCOLS + col];
    b.y = Wbig[(k0 + kh + 1) * YCOLS + col];
    acc = __builtin_amdgcn_wmma_f32_16x16x4_f32(false, a, false, b, (short)0, acc,
                                                false, false);
  }
  int rbase = rowTile * 16 + ((lane >> 4) << 3);
#pragma unroll
  for (int v = 0; v < 8; v++) {
    int r = rbase + v;
    if (r < nrows) Y[(size_t)r * YCOLS + col] = acc[v];
  }
}

// ---------------------------------------------------------- edge message
// 16 lanes per edge (2 edges per wave32). lane o:
//   hmid[o] = relu(b1[o] + ea[e,:]@W1[:,o]); broadcast hmid via __shfl,
//   msg[o]  = C[src,o] + sum_h hmid[h]*Y[src,h*16+o];  atomicAdd to agg[dst,o].
__global__ void edge_msg(const float* __restrict__ ea, const int* __restrict__ ei,
                         const float* __restrict__ W1, const float* __restrict__ b1,
                         const float* __restrict__ YA, float* __restrict__ agg, int nedges) {
  int t = blockIdx.x * blockDim.x + threadIdx.x;
  int e = t >> 4, o = t & 15;
  if (e >= nedges) return;
  int src = ei[e], dst = ei[nedges + e];
  const float* y = YA + (size_t)src * YCOLS;
  __builtin_prefetch(y, 0, 1);                 // global_prefetch of gathered Y row
  __builtin_prefetch(y + 144, 0, 1);
  float acc = b1[o];
#pragma unroll
  for (int k = 0; k < FE_IN; k++) acc = fmaf(ea[(size_t)e * FE_IN + k], W1[k * 16 + o], acc);
  float hm_self = fmaxf(acc, 0.0f);
  float msg = y[256 + o];                      // x@b2 term
  int base = threadIdx.x & 16;                 // 16-lane group within wave32
#pragma unroll
  for (int h = 0; h < 16; h++) {
    float hm = __shfl(hm_self, base | h, 32);
    msg = fmaf(hm, y[h * 16 + o], msg);
  }
  atomicAdd(&agg[(size_t)dst * 16 + o], msg);
}

// h_out = act(x@root + agg + bias); root term precomputed in YA cols 272..287
__global__ void node_update(const float* __restrict__ YA, const float* __restrict__ agg,
                            const float* __restrict__ bias, float* __restrict__ hout,
                            int nrows, int do_relu) {
  int t = blockIdx.x * blockDim.x + threadIdx.x;
  int n = t >> 4, o = t & 15;
  if (n >= nrows) return;
  float v = YA[(size_t)n * YCOLS + 272 + o] + agg[(size_t)n * 16 + o] + bias[o];
  hout[(size_t)n * 16 + o] = do_relu ? fmaxf(v, 0.0f) : v;
}

// ------------------------------------------------------------- mean pool
__global__ void mean_pool_acc(const float* __restrict__ h, const int* __restrict__ seg,
                              float* __restrict__ msum, float* __restrict__ cnt, int nrows) {
  int t = blockIdx.x * blockDim.x + threadIdx.x;
  int n = t >> 4, o = t & 15;
  if (n >= nrows) return;
  int b = seg[n];
  atomicAdd(&msum[(size_t)b * 16 + o], h[(size_t)n * 16 + o]);
  if (o == 0) atomicAdd(&cnt[b], 1.0f);
}

// -------------------------------------------------------- attention pool
// gate[n] = relu(x@W1+b1)@W2 + b2 ; one wave per node, lane j of 32
__global__ void att_gate(const float* __restrict__ x, const float* __restrict__ W1,
                         const float* __restrict__ b1, const float* __restrict__ W2,
                         const float* __restrict__ b2, float* __restrict__ gate, int nrows) {
  int n = blockIdx.x * (blockDim.x >> 5) + (threadIdx.x >> 5);
  if (n >= nrows) return;                      // wave-uniform
  int j = threadIdx.x & 31;
  const float* xr = x + (size_t)n * 32;
  float acc = b1[j];
#pragma unroll
  for (int i = 0; i < 32; i++) acc = fmaf(xr[i], W1[i * 32 + j], acc);
  float tv = fmaxf(acc, 0.0f) * W2[j];
#pragma unroll
  for (int off = 16; off > 0; off >>= 1) tv += __shfl_down(tv, off, 32);
  if (j == 0) gate[n] = tv + b2[0];
}

// monotone float->uint encoding for atomicMax
__global__ void seg_max_enc(const float* __restrict__ gate, const int* __restrict__ seg,
                            unsigned* __restrict__ m, int nrows) {
  int n = blockIdx.x * blockDim.x + threadIdx.x;
  if (n >= nrows) return;
  unsigned u = __float_as_uint(gate[n]);
  u = (u & 0x80000000u) ? ~u : (u | 0x80000000u);
  atomicMax(&m[seg[n]], u);
}

__global__ void exp_and_sum(float* __restrict__ gate, const int* __restrict__ seg,
                            const unsigned* __restrict__ m, float* __restrict__ esum,
                            int nrows) {
  int n = blockIdx.x * blockDim.x + threadIdx.x;
  if (n >= nrows) return;
  unsigned u = m[seg[n]];
  float md = (u & 0x80000000u) ? __uint_as_float(u & 0x7fffffffu) : __uint_as_float(~u);
  float ev = expf(gate[n] - md);
  gate[n] = ev;                                // reuse buffer for e
  atomicAdd(&esum[seg[n]], ev);
}

__global__ void att_acc(const float* __restrict__ en, const float* __restrict__ esum,
                        const float* __restrict__ x, const int* __restrict__ seg,
                        float* __restrict__ att, int nrows) {
  int n = blockIdx.x * (blockDim.x >> 5) + (threadIdx.x >> 5);
  if (n >= nrows) return;
  int j = threadIdx.x & 31;
  int b = seg[n];
  float a = en[n] / esum[b];
  atomicAdd(&att[(size_t)b * 32 + j], a * x[(size_t)n * 32 + j]);
}

// --------------------------------------------------------- final MLP head
// z = [mean_p(16) | mean_d(16) | att_p(32) | att_d(32)]; out = relu(z@W0+b0)@W1+b1
__global__ void final_head(const float* __restrict__ msum_p, const float* __restrict__ cnt_p,
                           const float* __restrict__ msum_d, const float* __restrict__ cnt_d,
                           const float* __restrict__ att_p, const float* __restrict__ att_d,
                           const float* __restrict__ W0, const float* __restrict__ b0,
                           const float* __restrict__ W1, const float* __restrict__ b1,
                           float* __restrict__ out, int nb) {
  int t = blockIdx.x * blockDim.x + threadIdx.x;
  int b = t >> 4, j = t & 15;
  if (b >= nb) return;
  float inv_p = 1.0f / fmaxf(cnt_p[b], 1.0f);
  float inv_d = 1.0f / fmaxf(cnt_d[b], 1.0f);
  float acc = b0[j];
#pragma unroll
  for (int k = 0; k < 16; k++) acc = fmaf(msum_p[b * 16 + k] * inv_p, W0[k * 16 + j], acc);
#pragma unroll
  for (int k = 0; k < 16; k++) acc = fmaf(msum_d[b * 16 + k] * inv_d, W0[(16 + k) * 16 + j], acc);
#pragma unroll
  for (int k = 0; k < 32; k++) acc = fmaf(att_p[b * 32 + k], W0[(32 + k) * 16 + j], acc);
#pragma unroll
  for (int k = 0; k < 32; k++) acc = fmaf(att_d[b * 32 + k], W0[(64 + k) * 16 + j], acc);
  float tv = fmaxf(acc, 0.0f) * W1[j];
#pragma unroll
  for (int off = 8; off > 0; off >>= 1) tv += __shfl_down(tv, off, 16);
  if (j == 0) out[b] = tv + b1[0];
}

// ---------------------------------------------------------------- driver
extern "C" void kernel_launch(void* const* d_in, const int* in_sizes, int n_in,
                              void* d_out, int out_size, void* d_ws, size_t ws_size,
                              hipStream_t stream) {
  (void)in_sizes; (void)n_in; (void)out_size; (void)ws_size;
  const float* x_p  = (const float*)d_in[0];
  const float* x_d  = (const float*)d_in[1];
  const float* ea_p = (const float*)d_in[2];
  const float* ea_d = (const float*)d_in[3];
  const float* pool_W1 = (const float*)d_in[4];
  const float* pool_b1 = (const float*)d_in[5];
  const float* pool_W2 = (const float*)d_in[6];
  const float* pool_b2 = (const float*)d_in[7];
  const float* lin0_W = (const float*)d_in[32];
  const float* lin0_b = (const float*)d_in[33];
  const float* lin1_W = (const float*)d_in[34];
  const float* lin1_b = (const float*)d_in[35];
  const int* ei_p = (const int*)d_in[36];
  const int* ei_d = (const int*)d_in[37];
  const int* batch_p = (const int*)d_in[38];
  const int* batch_d = (const int*)d_in[39];

  // workspace carve-out (~34 MB total)
  char* wsb = (char*)d_ws;
  size_t off = 0;
  auto carve = [&](size_t bytes) -> char* {
    char* p = wsb + off; off += (bytes + 255) & ~(size_t)255; return p;
  };
  float*    YA    = (float*)carve((size_t)N_NODES * YCOLS * 4);   // 28.8 MB
  float*    agg   = (float*)carve((size_t)N_NODES * 16 * 4);
  float*    hA    = (float*)carve((size_t)N_NODES * 16 * 4);
  float*    hB    = (float*)carve((size_t)N_NODES * 16 * 4);
  float*    Wbig  = (float*)carve((size_t)F_IN * YCOLS * 4);
  float*    gate  = (float*)carve((size_t)N_NODES * 4);
  unsigned* mEnc  = (unsigned*)carve((size_t)N_BATCH * 4);
  float*    esum  = (float*)carve((size_t)N_BATCH * 4);
  float*    attP  = (float*)carve((size_t)N_BATCH * 32 * 4);
  float*    attD  = (float*)carve((size_t)N_BATCH * 32 * 4);
  float*    msumP = (float*)carve((size_t)N_BATCH * 16 * 4);
  float*    msumD = (float*)carve((size_t)N_BATCH * 16 * 4);
  float*    cntP  = (float*)carve((size_t)N_BATCH * 4);
  float*    cntD  = (float*)carve((size_t)N_BATCH * 4);

  auto fz = [&](float* p, size_t n) {
    fill_zero_f32<<<(unsigned)((n + 255) / 256), 256, 0, stream>>>(p, (int)n);
  };

  const int rowTiles = (N_NODES + 15) / 16;
  const int gemmWaves = rowTiles * (YCOLS / 16);
  const unsigned gemmBlocks = (unsigned)((gemmWaves * 32 + 255) / 256);

  auto conv_layer = [&](const float* Xin, int cin, const float* W2w, const float* b2w,
                        const float* rootw, const float* W1w, const float* b1w,
                        const float* biasw, const float* eaw, const int* eiw,
                        float* hout, int do_relu) {
    pack_wbig<<<(unsigned)((cin * YCOLS + 255) / 256), 256, 0, stream>>>(W2w, b2w, rootw, Wbig, cin);
    wmma_gemm_f32<<<gemmBlocks, 256, 0, stream>>>(Xin, Wbig, YA, N_NODES, cin);
    fz(agg, (size_t)N_NODES * 16);
    edge_msg<<<(unsigned)((N_EDGES * 16) / 256), 256, 0, stream>>>(eaw, eiw, W1w, b1w, YA, agg, N_EDGES);
    node_update<<<(unsigned)((N_NODES * 16 + 255) / 256), 256, 0, stream>>>(YA, agg, biasw, hout, N_NODES, do_relu);
  };

  auto run_side = [&](int s, const float* x, const float* ea, const int* ei, const int* seg,
                      float* att, float* msum, float* cnt) {
    int pb = 8 + s * 12;
    const float* nn0_W1 = (const float*)d_in[pb + 0];   // (8,16)
    const float* nn0_b1 = (const float*)d_in[pb + 1];   // (16)
    const float* nn0_W2 = (const float*)d_in[pb + 2];   // (16,512)
    const float* nn0_b2 = (const float*)d_in[pb + 3];   // (512)
    const float* root0  = (const float*)d_in[pb + 4];   // (32,16)
    const float* bias0  = (const float*)d_in[pb + 5];   // (16)
    const float* nnR_W1 = (const float*)d_in[pb + 6];   // (2,8,16)
    const float* nnR_b1 = (const float*)d_in[pb + 7];   // (2,16)
    const float* nnR_W2 = (const float*)d_in[pb + 8];   // (2,16,256)
    const float* nnR_b2 = (const float*)d_in[pb + 9];   // (2,256)
    const float* rootR  = (const float*)d_in[pb + 10];  // (2,16,16)
    const float* biasR  = (const float*)d_in[pb + 11];  // (2,16)

    conv_layer(x, 32, nn0_W2, nn0_b2, root0, nn0_W1, nn0_b1, bias0, ea, ei, hA, 1);
    conv_layer(hA, 16, nnR_W2, nnR_b2, rootR, nnR_W1, nnR_b1, biasR, ea, ei, hB, 1);
    conv_layer(hB, 16, nnR_W2 + 4096, nnR_b2 + 256, rootR + 256,
               nnR_W1 + 128, nnR_b1 + 16, biasR + 16, ea, ei, hA, 0);

    fz(msum, (size_t)N_BATCH * 16);
    fz(cnt, (size_t)N_BATCH);
    mean_pool_acc<<<(unsigned)((N_NODES * 16 + 255) / 256), 256, 0, stream>>>(hA, seg, msum, cnt, N_NODES);

    att_gate<<<(unsigned)((N_NODES * 32 + 255) / 256), 256, 0, stream>>>(
        x, pool_W1, pool_b1, pool_W2, pool_b2, gate, N_NODES);
    fz((float*)mEnc, (size_t)N_BATCH);
    seg_max_enc<<<(unsigned)((N_NODES + 255) / 256), 256, 0, stream>>>(gate, seg, mEnc, N_NODES);
    fz(esum, (size_t)N_BATCH);
    exp_and_sum<<<(unsigned)((N_NODES + 255) / 256), 256, 0, stream>>>(gate, seg, mEnc, esum, N_NODES);
    fz(att, (size_t)N_BATCH * 32);
    att_acc<<<(unsigned)((N_NODES * 32 + 255) / 256), 256, 0, stream>>>(gate, esum, x, seg, att, N_NODES);
  };

  run_side(0, x_p, ea_p, ei_p, batch_p, attP, msumP, cntP);
  run_side(1, x_d, ea_d, ei_d, batch_d, attD, msumD, cntD);

  final_head<<<(unsigned)((N_BATCH * 16 + 255) / 256), 256, 0, stream>>>(
      msumP, cntP, msumD, cntD, attP, attD, lin0_W, lin0_b, lin1_W, lin1_b,
      (float*)d_out, N_BATCH);
}